// HypAgg_28114855919666
// MI455X (gfx1250) — compile-verified
//
#include <hip/hip_runtime.h>
#include <hip/hip_bf16.h>
#include <math.h>

// ---------------------------------------------------------------------------
// HypAgg (hyperbolic graph attention), B=2 N=512 D=128, all f32.
// - (relu(m) @ msg_w2) pulled out of the j-sum (linear), per-pair work O(D).
// - All GEMMs + the l_inner Gram matrix use V_WMMA_F32_16X16X4_F32 (exact f32).
// - Pair loop: one j per wave32, 4 k per lane -> pure shfl reductions, no
//   barriers in the hot loop (one __syncthreads per block at the end).
// ---------------------------------------------------------------------------

typedef __attribute__((ext_vector_type(2))) float v2f;
typedef __attribute__((ext_vector_type(8))) float v8f;

#define BN 1024      // B*N rows
#define NN 512       // nodes per batch
#define DD 128       // feature dim

__device__ __forceinline__ v8f wmma4(v2f a, v2f b, v8f c) {
  // 8 args: (neg_a, A, neg_b, B, c_mod, C, reuse_a, reuse_b)
  return __builtin_amdgcn_wmma_f32_16x16x4_f32(
      false, a, false, b, (short)0, c, false, false);
}

__device__ __forceinline__ float block_reduce_sum(float v, float* lds) {
  #pragma unroll
  for (int off = 16; off > 0; off >>= 1) v += __shfl_xor(v, off, 32);
  const int wave = threadIdx.x >> 5;
  const int lane = threadIdx.x & 31;
  __syncthreads();                 // protect lds from previous use
  if (lane == 0) lds[wave] = v;
  __syncthreads();
  float s = 0.0f;
  #pragma unroll
  for (int w = 0; w < 4; ++w) s += lds[w];
  return s;
}

// ---------------------------------------------------------------------------
// Stage 1: u = logmap0(x); xt = x with first component negated (so that
// dot(xt_i, x_j) == l_inner(x_i, x_j)).
// ---------------------------------------------------------------------------
__global__ void prep_kernel(const float* __restrict__ X,
                            float* __restrict__ U,
                            float* __restrict__ XT) {
  const int r = blockIdx.x;        // 0..1023
  const int k = threadIdx.x;       // 0..127
  const float* xr = X + r * DD;
  const float x0 = xr[0];
  const float z = fmaxf(x0, 1.0f + 1e-6f);
  const float dist = logf(z + sqrtf(z * z - 1.0f));          // arccosh(x0)
  const float coef = dist * rsqrtf(fmaxf(x0 * x0 - 1.0f, 1e-6f));
  const float xv = xr[k];
  U[r * DD + k]  = (k == 0) ? 0.0f : coef * xv;
  XT[r * DD + k] = (k == 0) ? -xv : xv;
}

// ---------------------------------------------------------------------------
// Generic (M x 128) @ (128 x 128) f32 WMMA GEMM, one wave per 16x16 tile.
// C = act(A@W + bias[n] + rowscale[m]*rbias[n])
// ---------------------------------------------------------------------------
__global__ void gemm128_kernel(const float* __restrict__ A,
                               const float* __restrict__ W,
                               float* __restrict__ C,
                               const float* __restrict__ bias,
                               const float* __restrict__ rowscale,
                               const float* __restrict__ rbias,
                               int do_relu) {
  const int lane = threadIdx.x;    // 0..31
  const int half = lane >> 4;
  const int l16  = lane & 15;
  const int tn = blockIdx.x;       // 0..7
  const int tm = blockIdx.y;       // 0..M/16-1
  const int arow = tm * 16 + l16;
  const int bcol = tn * 16 + l16;

  v8f acc = {};
  #pragma unroll 8
  for (int k0 = 0; k0 < DD; k0 += 4) {
    const float* ap = A + arow * DD + k0 + 2 * half;
    v2f a; a.x = ap[0]; a.y = ap[1];
    const float* wp = W + (k0 + 2 * half) * DD + bcol;
    v2f b; b.x = wp[0]; b.y = wp[DD];
    acc = wmma4(a, b, acc);
  }
  const float bn = bias ? bias[bcol] : 0.0f;
  const float rb = rbias ? rbias[bcol] : 0.0f;
  #pragma unroll
  for (int vr = 0; vr < 8; ++vr) {
    const int m = tm * 16 + half * 8 + vr;
    float v = acc[vr] + bn;
    if (rowscale) v += rowscale[m] * rb;
    if (do_relu) v = fmaxf(v, 0.0f);
    C[m * DD + bcol] = v;
  }
}

// ---------------------------------------------------------------------------
// Gram: G[b,i,j] = sum_k XT[b,i,k] * X[b,j,k]  ( == l_inner(x_i, x_j) )
// ---------------------------------------------------------------------------
__global__ void gram_kernel(const float* __restrict__ XT,
                            const float* __restrict__ X,
                            float* __restrict__ G) {
  const int lane = threadIdx.x;
  const int half = lane >> 4;
  const int l16  = lane & 15;
  const int tj = blockIdx.x;       // 0..31
  const int ti = blockIdx.y;       // 0..31
  const int b  = blockIdx.z;       // 0..1
  const float* Ab = XT + b * NN * DD;
  const float* Bb = X  + b * NN * DD;
  const int arow = ti * 16 + l16;
  const int bcol = tj * 16 + l16;

  v8f acc = {};
  #pragma unroll 8
  for (int k0 = 0; k0 < DD; k0 += 4) {
    const float* ap = Ab + arow * DD + k0 + 2 * half;
    v2f a; a.x = ap[0]; a.y = ap[1];
    const float* bp = Bb + bcol * DD + k0 + 2 * half;   // B[k][col] = X[col][k]
    v2f bb; bb.x = bp[0]; bb.y = bp[1];
    acc = wmma4(a, bb, acc);
  }
  float* Gb = G + b * NN * NN;
  #pragma unroll
  for (int vr = 0; vr < 8; ++vr) {
    const int i = ti * 16 + half * 8 + vr;
    Gb[i * NN + tj * 16 + l16] = acc[vr];
  }
}

// ---------------------------------------------------------------------------
// Stage 4: per row i: each of the 4 waves owns j = w, w+4, w+8, ...; each
// lane owns k = 4*lane..4*lane+3 (float4 coalesced loads). The D=128 dot for
// the attention logit is 4 lane-local FMAs + a 5-step shfl_xor wave reduce.
// No block barriers in the hot loop; 4 wave-partials combined once via LDS.
// ---------------------------------------------------------------------------
__global__ void pair_kernel(const float* __restrict__ ADJ,
                            const float* __restrict__ G,
                            const float* __restrict__ HI,
                            const float* __restrict__ HJ,
                            const float* __restrict__ MJ,
                            const float* __restrict__ att_w1,  // (258,128)
                            const float* __restrict__ att_b1,  // (128,)
                            const float* __restrict__ att_w2,  // (128,1)
                            const float* __restrict__ att_b2,  // (1,)
                            const float* __restrict__ msg_w1,  // (129,128)
                            const float* __restrict__ msg_b1,  // (128,)
                            float* __restrict__ ACC,
                            float* __restrict__ ATTSUM) {
  __shared__ float lacc[4 * DD];
  __shared__ float lsum[4];
  const int r = blockIdx.x;        // r = b*512 + i
  const int b = r >> 9;
  const int tid = threadIdx.x;     // 0..127
  const int w = tid >> 5;          // wave 0..3
  const int l = tid & 31;          // lane 0..31
  const int k0 = l * 4;            // this lane's 4 contiguous features

  const float4 hik  = *(const float4*)(HI + r * DD + k0);
  const float4 wadj = *(const float4*)(att_w1 + 256 * DD + k0);  // row 2d
  const float4 wedg = *(const float4*)(att_w1 + 257 * DD + k0);  // row 2d+1
  const float4 ab1  = *(const float4*)(att_b1 + k0);
  const float4 aw2  = *(const float4*)(att_w2 + k0);
  const float4 mwe  = *(const float4*)(msg_w1 + 128 * DD + k0);  // row d
  const float4 mb1  = *(const float4*)(msg_b1 + k0);
  const float b2 = att_b2[0];

  const float* adjr = ADJ + r * NN;
  const float* gr   = G + r * NN;
  const float* HJb  = HJ + (b << 9) * DD;
  const float* MJb  = MJ + (b << 9) * DD;

  float4 acc = make_float4(0.f, 0.f, 0.f, 0.f);
  float asum = 0.0f;
  for (int j = w; j < NN; j += 4) {
    const float adj = adjr[j];
    const float g = gr[j];
    const float zz = fmaxf(-g, 1.0f + 1e-6f);
    const float edge = logf(zz + sqrtf(zz * zz - 1.0f));   // arccosh(-l_inner)
    const float4 hj = *(const float4*)(HJb + j * DD + k0);
    const float4 mj = *(const float4*)(MJb + j * DD + k0);
    float p =
        fmaxf(hik.x + hj.x + adj * wadj.x + edge * wedg.x + ab1.x, 0.f) * aw2.x +
        fmaxf(hik.y + hj.y + adj * wadj.y + edge * wedg.y + ab1.y, 0.f) * aw2.y +
        fmaxf(hik.z + hj.z + adj * wadj.z + edge * wedg.z + ab1.z, 0.f) * aw2.z +
        fmaxf(hik.w + hj.w + adj * wadj.w + edge * wedg.w + ab1.w, 0.f) * aw2.w;
    #pragma unroll
    for (int off = 16; off > 0; off >>= 1) p += __shfl_xor(p, off, 32);
    const float att = 1.0f / (1.0f + expf(-(p + b2)));     // sigmoid
    acc.x += att * fmaxf(mj.x + edge * mwe.x + mb1.x, 0.f);
    acc.y += att * fmaxf(mj.y + edge * mwe.y + mb1.y, 0.f);
    acc.z += att * fmaxf(mj.z + edge * mwe.z + mb1.z, 0.f);
    acc.w += att * fmaxf(mj.w + edge * mwe.w + mb1.w, 0.f);
    asum += att;                                           // same in all lanes
  }
  *(float4*)(lacc + w * DD + k0) = acc;
  if (l == 0) lsum[w] = asum;
  __syncthreads();
  // tid == k: combine the 4 wave partials.
  const float s = lacc[tid] + lacc[DD + tid] + lacc[2 * DD + tid] + lacc[3 * DD + tid];
  ACC[r * DD + tid] = s;
  if (tid == 0) ATTSUM[r] = lsum[0] + lsum[1] + lsum[2] + lsum[3];
}

// ---------------------------------------------------------------------------
// Stage 6: msg[...,0]=0; transp0; expmap. One block per row.
// ---------------------------------------------------------------------------
__global__ void final_kernel(const float* __restrict__ X,
                             const float* __restrict__ MSG,
                             float* __restrict__ OUT) {
  __shared__ float lds[4];
  const int r = blockIdx.x;
  const int k = threadIdx.x;
  const float xv = X[r * DD + k];
  const float x0 = X[r * DD];                        // broadcast
  const float mv = (k == 0) ? 0.0f : MSG[r * DD + k];
  // l_inner(x, msg): msg0 == 0, so just the positive part of the dot.
  const float dot1 = block_reduce_sum(xv * mv, lds);
  const float coef = dot1 / (1.0f + x0);
  const float opx = (k == 0) ? (1.0f + x0) : xv;     // (o + x)_k
  const float uv = mv + coef * opx;                  // transp0 result
  float t = uv * uv;
  if (k == 0) t = -t;                                // Lorentz signature
  const float nu2 = block_reduce_sum(t, lds);
  const float nu = sqrtf(fmaxf(nu2, 1e-6f));
  const float ch = coshf(nu);
  const float sh = sinhf(nu) / nu;
  OUT[r * DD + k] = ch * xv + sh * uv;
}

// ---------------------------------------------------------------------------
extern "C" void kernel_launch(void* const* d_in, const int* in_sizes, int n_in,
                              void* d_out, int out_size, void* d_ws, size_t ws_size,
                              hipStream_t stream) {
  const float* x      = (const float*)d_in[0];
  const float* adj    = (const float*)d_in[1];
  const float* att_w1 = (const float*)d_in[2];
  const float* att_b1 = (const float*)d_in[3];
  const float* att_w2 = (const float*)d_in[4];
  const float* att_b2 = (const float*)d_in[5];
  const float* msg_w1 = (const float*)d_in[6];
  const float* msg_b1 = (const float*)d_in[7];
  const float* msg_w2 = (const float*)d_in[8];
  const float* msg_b2 = (const float*)d_in[9];
  const float* out_w1 = (const float*)d_in[10];
  const float* out_b1 = (const float*)d_in[11];
  const float* out_w2 = (const float*)d_in[12];
  const float* out_b2 = (const float*)d_in[13];

  float* ws = (float*)d_ws;
  const size_t RD = (size_t)BN * DD;      // 131072
  float* U      = ws;
  float* XT     = U + RD;
  float* HI     = XT + RD;
  float* HJ     = HI + RD;
  float* MJ     = HJ + RD;
  float* ACC    = MJ + RD;
  float* MSG1   = ACC + RD;
  float* TB     = MSG1 + RD;
  float* MSG2   = TB + RD;
  float* ATTSUM = MSG2 + RD;              // 1024
  float* GRAM   = ATTSUM + BN;            // 524288

  prep_kernel<<<BN, DD, 0, stream>>>(x, U, XT);

  dim3 gemm_grid(8, BN / 16);             // N-tiles x M-tiles
  gemm128_kernel<<<gemm_grid, 32, 0, stream>>>(U, att_w1,            HI, nullptr, nullptr, nullptr, 0);
  gemm128_kernel<<<gemm_grid, 32, 0, stream>>>(U, att_w1 + DD * DD,  HJ, nullptr, nullptr, nullptr, 0);
  gemm128_kernel<<<gemm_grid, 32, 0, stream>>>(U, msg_w1,            MJ, nullptr, nullptr, nullptr, 0);

  gram_kernel<<<dim3(NN / 16, NN / 16, 2), 32, 0, stream>>>(XT, x, GRAM);

  pair_kernel<<<BN, DD, 0, stream>>>(adj, GRAM, HI, HJ, MJ,
                                     att_w1, att_b1, att_w2, att_b2,
                                     msg_w1, msg_b1, ACC, ATTSUM);

  // msg1 = ACC @ msg_w2 + attsum[m] * msg_b2[n]
  gemm128_kernel<<<gemm_grid, 32, 0, stream>>>(ACC,  msg_w2, MSG1, nullptr, ATTSUM, msg_b2, 0);
  // t = relu(msg1 @ out_w1 + out_b1)
  gemm128_kernel<<<gemm_grid, 32, 0, stream>>>(MSG1, out_w1, TB,   out_b1, nullptr, nullptr, 1);
  // msg2 = t @ out_w2 + out_b2
  gemm128_kernel<<<gemm_grid, 32, 0, stream>>>(TB,   out_w2, MSG2, out_b2, nullptr, nullptr, 0);

  final_kernel<<<BN, DD, 0, stream>>>(x, MSG2, (float*)d_out);
}